// MultipleCandidateAttention_18056042512910
// MI455X (gfx1250) — compile-verified
//
#include <hip/hip_runtime.h>
#include <hip/hip_bf16.h>

typedef float v2f __attribute__((ext_vector_type(2)));
typedef float v8f __attribute__((ext_vector_type(8)));

// gfx1250 native V_TANH_F32 (TRANS pipe).  Prefer the clang builtin (best
// scheduling); otherwise emit the instruction directly via inline asm — the
// ISA lists TANH among gfx1250 TRANS32 ops, so the mnemonic must assemble.
static __device__ __forceinline__ float fast_tanh(float x) {
#if __has_builtin(__builtin_amdgcn_tanhf)
    return __builtin_amdgcn_tanhf(x);
#else
    float r;
    asm("v_tanh_f32 %0, %1" : "=v"(r) : "v"(x));
    return r;
#endif
}

// ---------------------------------------------------------------------------
// Batched fp32 WMMA GEMM:  C[b] = A[b] * B[b] (+ bias per column).
// One wave computes a 16 x (16*NT) strip of C: a single A fragment (16x4)
// feeds NT back-to-back V_WMMA_F32_16X16X4_F32 ops, amortizing A loads and
// address math.  M % 16 == 0, N % (16*NT) == 0, K % 4 == 0 for all uses here.
//
// VGPR layouts per CDNA5 ISA 7.12.2:
//   A 16x4: lanes 0-15 -> M=lane,   (a.x,a.y)=(K+0,K+1)
//           lanes16-31 -> M=lane-16,(a.x,a.y)=(K+2,K+3)
//   B 4x16 mirrored: lanes 0-15 -> N=lane, (b.x,b.y)=(K+0,K+1); hi half K+2/3
//   C/D 16x16 f32: VGPR v -> M=v (lanes 0-15) / v+8 (lanes 16-31), N=lane&15
// ---------------------------------------------------------------------------
template <int NT>
__global__ void wmma_gemm_f32(const float* __restrict__ A,
                              const float* __restrict__ Bm,
                              const float* __restrict__ bias,
                              float* __restrict__ C,
                              int M, int N, int K,
                              size_t sA, size_t sB, size_t sC)
{
    const int batch = blockIdx.z;
    A  += (size_t)batch * sA;
    Bm += (size_t)batch * sB;
    C  += (size_t)batch * sC;

    const int tN   = blockIdx.x * 16 * NT;
    const int tM   = blockIdx.y * 16;
    const int lane = threadIdx.x & 31;
    const int half = lane >> 4;    // 0: lanes 0-15, 1: lanes 16-31
    const int l16  = lane & 15;

    const float* __restrict__ aRow = A + (size_t)(tM + l16) * K;

    v8f acc[NT];
    #pragma unroll
    for (int t = 0; t < NT; ++t) acc[t] = (v8f){};

    #pragma unroll 2
    for (int k = 0; k < K; k += 4) {
        const int ka = k + half * 2;
        v2f av;
        av.x = aRow[ka];                       // merges into global_load_b64
        av.y = aRow[ka + 1];

        const float* __restrict__ b0 = Bm + (size_t)ka * N + tN + l16;
        const float* __restrict__ b1 = b0 + N;

        #pragma unroll
        for (int t = 0; t < NT; ++t) {
            v2f bv;
            bv.x = b0[t * 16];
            bv.y = b1[t * 16];
            acc[t] = __builtin_amdgcn_wmma_f32_16x16x4_f32(
                /*neg_a=*/false, av, /*neg_b=*/false, bv,
                /*c_mod=*/(short)0, acc[t], /*reuse=*/false, false);
        }
    }

    #pragma unroll
    for (int t = 0; t < NT; ++t) {
        const int col  = tN + t * 16 + l16;
        const float bv = bias ? bias[col] : 0.0f;
        #pragma unroll
        for (int v = 0; v < 8; ++v) {
            const int m = tM + v + half * 8;
            C[(size_t)m * N + col] = acc[t][v] + bv;
        }
    }
}

// ---------------------------------------------------------------------------
// Scores: a[b,q,f] = sum_a tanh(pf[b,f,a] + pq[b,q,a]) * Wa[a], masked.
// One wave per (b,q,f); each lane handles 8 contiguous floats (2x float4 ->
// global_load_b128) so the 67M-tanh TRANS workload streams at full rate,
// then a 32-lane shuffle reduction.
// ---------------------------------------------------------------------------
__global__ void scores_kernel(const float* __restrict__ pf,
                              const float* __restrict__ pq,
                              const float* __restrict__ Wa,
                              const int* __restrict__ mask,
                              float* __restrict__ out)
{
    const int wavesPerBlock = blockDim.x >> 5;
    const int widx = blockIdx.x * wavesPerBlock + (threadIdx.x >> 5);
    const int lane = threadIdx.x & 31;

    const int f = widx & 127;          // F = 128
    const int q = (widx >> 7) & 31;    // Q = 32
    const int b = widx >> 12;          // B index

    const float4* pf4 = (const float4*)(pf + ((size_t)(b * 128 + f)) * 256);
    const float4* pq4 = (const float4*)(pq + ((size_t)(b * 32 + q)) * 256);
    const float4* wa4 = (const float4*)Wa;

    float s = 0.0f;
    #pragma unroll
    for (int i = 0; i < 2; ++i) {
        const int idx = lane * 2 + i;                  // 64 float4 = 256 floats
        const float4 x = pf4[idx];
        const float4 y = pq4[idx];
        const float4 w = wa4[idx];
        s += fast_tanh(x.x + y.x) * w.x;
        s += fast_tanh(x.y + y.y) * w.y;
        s += fast_tanh(x.z + y.z) * w.z;
        s += fast_tanh(x.w + y.w) * w.w;
    }
    #pragma unroll
    for (int off = 16; off > 0; off >>= 1)
        s += __shfl_down(s, off, 32);

    if (lane == 0) {
        out[widx] = (mask[b * 128 + f] == 0) ? -1e9f : s;
    }
}

// ---------------------------------------------------------------------------
// Softmax over F=128, one wave per (b,q) row, 4 elements/lane via float4.
// Operates in place on the score buffer.
// ---------------------------------------------------------------------------
__global__ void softmax_kernel(float* __restrict__ a)
{
    const int wavesPerBlock = blockDim.x >> 5;
    const int row  = blockIdx.x * wavesPerBlock + (threadIdx.x >> 5); // b*32+q
    const int lane = threadIdx.x & 31;

    float4* row4 = (float4*)(a + (size_t)row * 128);
    const float4 v = row4[lane];

    float m = fmaxf(fmaxf(v.x, v.y), fmaxf(v.z, v.w));
    #pragma unroll
    for (int off = 16; off > 0; off >>= 1)
        m = fmaxf(m, __shfl_xor(m, off, 32));

    float4 e;
    e.x = __expf(v.x - m);
    e.y = __expf(v.y - m);
    e.z = __expf(v.z - m);
    e.w = __expf(v.w - m);

    float s = e.x + e.y + e.z + e.w;
    #pragma unroll
    for (int off = 16; off > 0; off >>= 1)
        s += __shfl_xor(s, off, 32);

    const float inv = 1.0f / s;
    float4 o;
    o.x = e.x * inv; o.y = e.y * inv; o.z = e.z * inv; o.w = e.w * inv;
    row4[lane] = o;
}

// ---------------------------------------------------------------------------
// Launcher.  Shapes: B=64, F=128, Q=32, Df=Dq=400, A=256.
// Workspace: pf [8192x256] f32 (8 MB) | pq [2048x256] f32 (2 MB) |
//            scores/alpha [64x32x128] f32 (1 MB, softmax in place).
// ---------------------------------------------------------------------------
extern "C" void kernel_launch(void* const* d_in, const int* in_sizes, int n_in,
                              void* d_out, int out_size, void* d_ws, size_t ws_size,
                              hipStream_t stream)
{
    const float* feature = (const float*)d_in[0];   // [64,128,400]
    const float* query   = (const float*)d_in[1];   // [64,32,400]
    const int*   mask    = (const int*)  d_in[2];   // [64,128]
    const float* Wf      = (const float*)d_in[3];   // [400,256]
    const float* Wq      = (const float*)d_in[4];   // [400,256]
    const float* bq      = (const float*)d_in[5];   // [256]
    const float* Wa      = (const float*)d_in[6];   // [256]
    float*       out     = (float*)d_out;           // [64,32,400]

    char* ws = (char*)d_ws;
    float* pf = (float*)ws;                                   // 8192*256 f32
    float* pq = (float*)(ws + (size_t)8192 * 256 * 4);        // 2048*256 f32
    float* sc = (float*)(ws + (size_t)8192 * 256 * 4
                            + (size_t)2048 * 256 * 4);        // 262144 f32

    const dim3 waveBlk(32);

    // pf = feature @ Wf : M=8192, N=256, K=400 ; NT=4 -> 4 x 512 strip grid
    wmma_gemm_f32<4><<<dim3(256 / 64, 8192 / 16, 1), waveBlk, 0, stream>>>(
        feature, Wf, nullptr, pf, 8192, 256, 400, 0, 0, 0);

    // pq = query @ Wq + bq : M=2048, N=256, K=400 ; NT=4
    wmma_gemm_f32<4><<<dim3(256 / 64, 2048 / 16, 1), waveBlk, 0, stream>>>(
        query, Wq, bq, pq, 2048, 256, 400, 0, 0, 0);

    // scores + mask : 64*32*128 = 262144 waves, 8 waves per 256-thread block
    scores_kernel<<<262144 / 8, 256, 0, stream>>>(pf, pq, Wa, mask, sc);

    // softmax over F : 2048 rows, 8 waves per block
    softmax_kernel<<<2048 / 8, 256, 0, stream>>>(sc);

    // out[b] = alpha[b] @ feature[b] : M=32, N=400, K=128, batch=64 ; NT=5
    wmma_gemm_f32<5><<<dim3(400 / 80, 32 / 16, 64), waveBlk, 0, stream>>>(
        sc, feature, nullptr, out, 32, 400, 128,
        (size_t)32 * 128, (size_t)128 * 400, (size_t)32 * 400);
}